// SynapticPlasticity_51290499449084
// MI455X (gfx1250) — compile-verified
//
#include <hip/hip_runtime.h>
#include <hip/hip_bf16.h>

typedef __attribute__((ext_vector_type(2))) float v2f;
typedef __attribute__((ext_vector_type(8))) float v8f;

#define TAU_PRE  20.0f
#define TAU_POST 20.0f
#define LR       0.01f

#define BB 16
#define SS 8192
#define II 256
#define HH 256
#define M_TOTAL (BB * SS)        /* 131072 flattened (b,s) rows */
#define NSPLIT  32               /* split-K factor for dw reduction */

/* dw kernel LDS slab geometry: rows padded so the two 16-lane halves of a
 * wave (which read rows r and r+2 / cols 0..15) hit disjoint bank groups. */
#define PAD_H 136                /* 128-wide h slabs, stride 136 floats */
#define PAD_I 72                 /* 64-wide  i slabs, stride 72 floats  */
#define SP_OFF 0                 /* post_spikes slab  [32][136]          */
#define PT_OFF (32 * PAD_H)      /* post_tr slab      [32][136]          */
#define TR_OFF (2 * 32 * PAD_H)  /* pre_tr slab       [32][72]           */
#define PS_OFF (TR_OFF + 32 * PAD_I) /* pre_spikes slab [32][72]         */
#define LDS_FLOATS (PS_OFF + 32 * PAD_I) /* 13312 floats = 53 KB/buffer  */

// -------------------------------------------------------------------------
// CDNA5 async global->LDS copy (ASYNCcnt-tracked), 16 bytes per lane.
// -------------------------------------------------------------------------
__device__ __forceinline__ void async_copy_b128(const float* lds_dst,
                                                const float* gsrc) {
    unsigned lds_addr = (unsigned)(unsigned long long)lds_dst;
    asm volatile("global_load_async_to_lds_b128 %0, %1, off"
                 :: "v"(lds_addr), "v"(gsrc)
                 : "memory");
}

__device__ __forceinline__ void wait_asynccnt0() {
    asm volatile("s_wait_asynccnt 0x0" ::: "memory");
}

// -------------------------------------------------------------------------
// Kernel 1: trace decay + spike accumulation (elementwise, float4 streams)
// -------------------------------------------------------------------------
__global__ void stdp_traces_kernel(const float4* __restrict__ pre_s,
                                   const float4* __restrict__ post_s,
                                   const float4* __restrict__ pre_t,
                                   const float4* __restrict__ post_t,
                                   const int* __restrict__ dt,
                                   float4* __restrict__ pre_tr,
                                   float4* __restrict__ post_tr) {
    const float dtf = (float)dt[0];
    const float dp  = expf(-dtf / TAU_PRE);
    const float dq  = expf(-dtf / TAU_POST);
    const size_t i = (size_t)blockIdx.x * blockDim.x + threadIdx.x;

    float4 a = pre_t[i];
    float4 s = pre_s[i];
    float4 r;
    r.x = fmaf(a.x, dp, s.x); r.y = fmaf(a.y, dp, s.y);
    r.z = fmaf(a.z, dp, s.z); r.w = fmaf(a.w, dp, s.w);
    pre_tr[i] = r;

    a = post_t[i];
    s = post_s[i];
    r.x = fmaf(a.x, dq, s.x); r.y = fmaf(a.y, dq, s.y);
    r.z = fmaf(a.z, dq, s.z); r.w = fmaf(a.w, dq, s.w);
    post_tr[i] = r;
}

// -------------------------------------------------------------------------
// Kernel 2: output = pre_spikes @ W^T via V_WMMA_F32_16X16X4_F32.
// One wave computes a 16(m) x 64(h) strip (4 n-tiles) so each A fragment is
// reused across 4 WMMAs: 5 b64 loads per 4 WMMAs, 4x less A re-read.
// -------------------------------------------------------------------------
__global__ void stdp_output_gemm_kernel(const float* __restrict__ A,  // [M,II]
                                        const float* __restrict__ W,  // [HH,II]
                                        float* __restrict__ Out) {    // [M,HH]
    const int wid   = blockIdx.x * (blockDim.x >> 5) + (threadIdx.x >> 5);
    const int lane  = threadIdx.x & 31;
    const int nquad = wid & 3;           // 4 groups of 64 columns over HH
    const int mtile = wid >> 2;          // 8192 m-tiles
    const int lid   = lane & 15;
    const int kb    = (lane >> 4) * 2;

    const float* aptr = A + (size_t)(mtile * 16 + lid) * II + kb;
    const float* bptr = W + (size_t)(nquad * 64 + lid) * II + kb;

    v8f c0 = {}, c1 = {}, c2 = {}, c3 = {};
#pragma unroll 4
    for (int k = 0; k < II; k += 4) {
        v2f a  = *(const v2f*)(aptr + k);
        v2f b0 = *(const v2f*)(bptr + k);
        v2f b1 = *(const v2f*)(bptr + 16 * II + k);
        v2f b2 = *(const v2f*)(bptr + 32 * II + k);
        v2f b3 = *(const v2f*)(bptr + 48 * II + k);
        c0 = __builtin_amdgcn_wmma_f32_16x16x4_f32(false, a, false, b0, (short)0, c0, false, false);
        c1 = __builtin_amdgcn_wmma_f32_16x16x4_f32(false, a, false, b1, (short)0, c1, false, false);
        c2 = __builtin_amdgcn_wmma_f32_16x16x4_f32(false, a, false, b2, (short)0, c2, false, false);
        c3 = __builtin_amdgcn_wmma_f32_16x16x4_f32(false, a, false, b3, (short)0, c3, false, false);
    }

    const int mbase = mtile * 16 + ((lane >> 4) * 8);
#pragma unroll
    for (int r = 0; r < 8; ++r) {
        const size_t rowoff = (size_t)(mbase + r) * HH + nquad * 64 + lid;
        Out[rowoff +  0] = c0[r];
        Out[rowoff + 16] = c1[r];
        Out[rowoff + 32] = c2[r];
        Out[rowoff + 48] = c3[r];
    }
}

// -------------------------------------------------------------------------
// Async staging of one 32-m chunk into an LDS buffer (coalesced b128/lane).
// -------------------------------------------------------------------------
__device__ __forceinline__ void stage_chunk(float* shbuf, int t,
                                            const float* __restrict__ postS,
                                            const float* __restrict__ postTR,
                                            const float* __restrict__ preTR,
                                            const float* __restrict__ preS,
                                            int mbase, int h0, int i0) {
#pragma unroll
    for (int j = 0; j < 4; ++j) {        // h slabs: 1024 float4 slots each
        const int s  = t + j * 256;
        const int r  = s >> 5;           // 32 float4 per 128-wide row
        const int c4 = (s & 31) * 4;
        const size_t g = (size_t)(mbase + r) * 256 + h0 + c4;
        async_copy_b128(shbuf + SP_OFF + r * PAD_H + c4, postS  + g);
        async_copy_b128(shbuf + PT_OFF + r * PAD_H + c4, postTR + g);
    }
#pragma unroll
    for (int j = 0; j < 2; ++j) {        // i slabs: 512 float4 slots each
        const int s  = t + j * 256;
        const int r  = s >> 4;           // 16 float4 per 64-wide row
        const int c4 = (s & 15) * 4;
        const size_t g = (size_t)(mbase + r) * 256 + i0 + c4;
        async_copy_b128(shbuf + TR_OFF + r * PAD_I + c4, preTR + g);
        async_copy_b128(shbuf + PS_OFF + r * PAD_I + c4, preS  + g);
    }
}

// -------------------------------------------------------------------------
// Kernel 3: split-K dw partials, double-buffered async-LDS staging.
// dw[h,i] = sum_m ( postS[m,h]*preTR[m,i] - postTR[m,h]*preS[m,i] )
// Block = 128(h) x 64(i) macro-tile.  While waves compute chunk c from one
// LDS buffer, chunk c+1 streams into the other via ASYNCcnt-tracked
// global_load_async_to_lds_b128.  Wave w owns the h = h0+w*16 strip, so
// split partials are written directly (deterministic, no atomics).
// -------------------------------------------------------------------------
__global__ void stdp_dw_gemm_kernel(const float* __restrict__ postS,  // [M,HH]
                                    const float* __restrict__ preTR,  // [M,II]
                                    const float* __restrict__ postTR, // [M,HH]
                                    const float* __restrict__ preS,   // [M,II]
                                    float* __restrict__ partial) {    // [NSPLIT,HH,II]
    __shared__ float sh[2][LDS_FLOATS];  // ~104 KB of the 320 KB/WGP

    const int macro = blockIdx.x;            // 0..7 : 2 (h) x 4 (i)
    const int split = blockIdx.y;            // 0..NSPLIT-1
    const int h0 = (macro >> 2) * 128;
    const int i0 = (macro & 3) * 64;

    const int t    = threadIdx.x;
    const int wave = t >> 5;
    const int lane = t & 31;
    const int lid  = lane & 15;
    const int kb   = (lane >> 4) * 2;
    const int hcol = wave * 16 + lid;        // column in 128-wide h slabs

    const int mPerSplit = M_TOTAL / NSPLIT;  // 4096
    const int mBlock0   = split * mPerSplit;
    const int nChunks   = mPerSplit / 32;    // 128

    v8f accP[4] = {{}, {}, {}, {}};
    v8f accN[4] = {{}, {}, {}, {}};

    // prologue: stage chunk 0
    stage_chunk(sh[0], t, postS, postTR, preTR, preS, mBlock0, h0, i0);

    for (int c = 0; c < nChunks; ++c) {
        wait_asynccnt0();        // my chunk-c transfers have landed in LDS
        __syncthreads();         // all waves' transfers landed; previous
                                 // readers of the buffer staged next are done

        if (c + 1 < nChunks)
            stage_chunk(sh[(c + 1) & 1], t, postS, postTR, preTR, preS,
                        mBlock0 + (c + 1) * 32, h0, i0);

        const float* buf = sh[c & 1];

        // ---- compute: 8 k-steps of 4 m-rows, 8 WMMAs each ----------------
#pragma unroll
        for (int kq = 0; kq < 8; ++kq) {
            const int row = kq * 4 + kb;
            v2f a1, a2;
            a1.x = buf[SP_OFF + (row    ) * PAD_H + hcol];
            a1.y = buf[SP_OFF + (row + 1) * PAD_H + hcol];
            a2.x = buf[PT_OFF + (row    ) * PAD_H + hcol];
            a2.y = buf[PT_OFF + (row + 1) * PAD_H + hcol];
#pragma unroll
            for (int it = 0; it < 4; ++it) {
                const int icol = it * 16 + lid;
                v2f b1, b2;
                b1.x = buf[TR_OFF + (row    ) * PAD_I + icol];
                b1.y = buf[TR_OFF + (row + 1) * PAD_I + icol];
                b2.x = buf[PS_OFF + (row    ) * PAD_I + icol];
                b2.y = buf[PS_OFF + (row + 1) * PAD_I + icol];
                accP[it] = __builtin_amdgcn_wmma_f32_16x16x4_f32(
                               false, a1, false, b1, (short)0, accP[it], false, false);
                accN[it] = __builtin_amdgcn_wmma_f32_16x16x4_f32(
                               false, a2, false, b2, (short)0, accN[it], false, false);
            }
        }
        __syncthreads();         // seal reads of buf before it is restaged
    }

    // ---- write this wave's 16x64 strip of the split partial --------------
    const int hbase = h0 + wave * 16 + ((lane >> 4) * 8);
#pragma unroll
    for (int it = 0; it < 4; ++it) {
        const int i = i0 + it * 16 + lid;
#pragma unroll
        for (int r = 0; r < 8; ++r) {
            partial[(size_t)split * (HH * II) + (size_t)(hbase + r) * II + i] =
                accP[it][r] - accN[it][r];
        }
    }
}

// -------------------------------------------------------------------------
// Kernel 4: new_w = clip(w + (sum_splits partial) * LR / n, -1, 1)
// -------------------------------------------------------------------------
__global__ void stdp_finalize_kernel(const float* __restrict__ w,
                                     const float* __restrict__ partial,
                                     float* __restrict__ neww) {
    const int j = blockIdx.x * blockDim.x + threadIdx.x;
    float s = 0.0f;
#pragma unroll
    for (int sp = 0; sp < NSPLIT; ++sp) s += partial[(size_t)sp * (HH * II) + j];
    const float scale = LR / (float)M_TOTAL;
    float nw = fmaf(s, scale, w[j]);
    nw = fminf(1.0f, fmaxf(-1.0f, nw));
    neww[j] = nw;
}

// -------------------------------------------------------------------------
extern "C" void kernel_launch(void* const* d_in, const int* in_sizes, int n_in,
                              void* d_out, int out_size, void* d_ws, size_t ws_size,
                              hipStream_t stream) {
    const float* pre_spikes  = (const float*)d_in[0];  // [B,S,I]
    const float* post_spikes = (const float*)d_in[1];  // [B,S,H]
    const float* pre_trace   = (const float*)d_in[2];  // [B,S,I]
    const float* post_trace  = (const float*)d_in[3];  // [B,S,H]
    const float* weights     = (const float*)d_in[4];  // [H,I]
    const int*   dt          = (const int*)d_in[5];    // scalar

    float* out = (float*)d_out;
    const size_t n_big = (size_t)M_TOTAL * HH;         // 33554432 (== M*II)
    float* output  = out;                              // [B,S,H]
    float* new_w   = out + n_big;                      // [H,I]
    float* pre_tr  = new_w + (size_t)HH * II;          // [B,S,I]
    float* post_tr = pre_tr + n_big;                   // [B,S,H]

    float* partial = (float*)d_ws;                     // NSPLIT*65536 floats = 8 MB

    // 1) traces (elementwise); must precede dw GEMM (which reads pre_tr/post_tr)
    {
        const size_t n4 = n_big / 4;                   // 8388608 float4 per tensor
        dim3 grid((unsigned)(n4 / 256));
        stdp_traces_kernel<<<grid, 256, 0, stream>>>(
            (const float4*)pre_spikes, (const float4*)post_spikes,
            (const float4*)pre_trace,  (const float4*)post_trace,
            dt, (float4*)pre_tr, (float4*)post_tr);
    }

    // 2) output GEMM: 8192 m-tiles x 4 n-quads = 32768 waves / 8 = 4096 blocks
    {
        dim3 grid((M_TOTAL / 16) * (HH / 64) / 8);
        stdp_output_gemm_kernel<<<grid, 256, 0, stream>>>(pre_spikes, weights, output);
    }

    // 3) dw split-K partials: 8 macro-tiles x NSPLIT splits
    {
        dim3 grid(8, NSPLIT);
        stdp_dw_gemm_kernel<<<grid, 256, 0, stream>>>(
            post_spikes, pre_tr, post_tr, pre_spikes, partial);
    }

    // 4) finalize weights
    {
        dim3 grid((HH * II) / 256);
        stdp_finalize_kernel<<<grid, 256, 0, stream>>>(weights, partial, new_w);
    }
}